// MaximumMeanDiscrepancy_19138374271586
// MI455X (gfx1250) — compile-verified
//
#include <hip/hip_runtime.h>

// MMD-RBF fused kernel for MI455X (gfx1250, wave32, WMMA).
// B=4, Ns=Nt=2048, K=32.  Pass1: bandwidth via closed-form sum(L2).
// Pass2: 64x64 upper-tri tiles, V_WMMA_F32_16X16X4_F32 Gram + fused 5-exp RBF.
// Pass3: fixed-order quadrant reduction (deterministic, no float atomics).
// Workspace: 32 floats (per-batch exp constants) + 4*64*64 tile sums = ~66KB.

#define KDIM   32
#define NSIDE  2048
#define NTOT   4096
#define TILE   64
#define TGRID  64          // NTOT / TILE
#define QTILES 32          // NSIDE / TILE  (quadrant boundary in tiles)
#define NBATCH 4
#define LOG2E  1.4426950408889634f

typedef float v2f __attribute__((ext_vector_type(2)));
typedef float v8f __attribute__((ext_vector_type(8)));

__device__ __forceinline__ const float* row_ptr(const float* __restrict__ src,
                                                const float* __restrict__ tgt,
                                                int b, int r) {
  return (r < NSIDE) ? src + ((size_t)b * NSIDE + r) * KDIM
                     : tgt + ((size_t)b * NSIDE + (r - NSIDE)) * KDIM;
}

// ---------------- Pass 1: bandwidth constants ----------------
// sum(L2) = 2N*sum(|x|^2) - 2*|sum(x)|^2  -> bw -> c_i = -log2e / bw_i
__global__ void mmd_pass1(const float* __restrict__ src,
                          const float* __restrict__ tgt,
                          float* __restrict__ ci) {
  const int b = blockIdx.x;
  const int t = threadIdx.x;       // 256 threads
  const int k = t & 31;            // column 0..31 (coalesced)
  const int g = t >> 5;            // row group 0..7
  float sacc = 0.f, qacc = 0.f;
  for (int r = g; r < NTOT; r += 8) {
    float v = row_ptr(src, tgt, b, r)[k];
    sacc += v;
    qacc += v * v;
  }
  __shared__ float partS[8][32];
  __shared__ float partQ[256];
  partS[g][k] = sacc;
  partQ[t]    = qacc;
  __syncthreads();
  for (int off = 128; off > 0; off >>= 1) {
    if (t < off) partQ[t] += partQ[t + off];
    __syncthreads();
  }
  if (t == 0) {
    float s2 = 0.f;
    for (int kk = 0; kk < 32; ++kk) {
      float s = 0.f;
      for (int gg = 0; gg < 8; ++gg) s += partS[gg][kk];
      s2 += s * s;
    }
    const float nf    = (float)NTOT;
    const float sumL2 = 2.f * nf * partQ[0] - 2.f * s2;
    float bw = sumL2 / (nf * nf - nf + 1e-8f);
    bw *= 0.25f;                      // / KERNEL_MUL^(KERNEL_NUM/2) = / 2^2
    for (int i = 0; i < 5; ++i) {
      ci[b * 8 + i] = -LOG2E / bw;    // exp(-L2/bw_i) = exp2(L2*ci + d)
      bw *= 2.0f;                     // KERNEL_MUL
    }
  }
}

// ---------------- Pass 2: fused WMMA Gram + RBF kernel tile sums ----------------
__device__ __forceinline__ float kern5(float L2, float c0, float c1, float c2,
                                       float c3, float c4) {
  const float d = 1e-8f * LOG2E;      // the +1e-8 inside exp, in log2 domain
  float s;
  s  = __builtin_amdgcn_exp2f(fmaf(L2, c0, d));
  s += __builtin_amdgcn_exp2f(fmaf(L2, c1, d));
  s += __builtin_amdgcn_exp2f(fmaf(L2, c2, d));
  s += __builtin_amdgcn_exp2f(fmaf(L2, c3, d));
  s += __builtin_amdgcn_exp2f(fmaf(L2, c4, d));
  return s;
}

__global__ __launch_bounds__(128)
void mmd_pass2(const float* __restrict__ src, const float* __restrict__ tgt,
               const float* __restrict__ ci, float* __restrict__ tileSum) {
  const int tj = blockIdx.x;
  const int ti = blockIdx.y;
  const int b  = blockIdx.z;
  if (ti > tj) return;               // upper triangle only (uniform exit)

  __shared__ float ldsX[TILE][KDIM + 2];   // stride 34: conflict-free, 8B-aligned
  __shared__ float ldsY[TILE][KDIM + 2];
  __shared__ float sqX[TILE], sqY[TILE];
  __shared__ float redBuf[4];

  const int t = threadIdx.x;
  {  // cooperative tile load: thread t<64 -> X row t, t>=64 -> Y row t-64
    const int  local = t & 63;
    const bool isY   = t >= 64;
    const int  gr    = (isY ? tj : ti) * TILE + local;
    const float4* p  = (const float4*)row_ptr(src, tgt, b, gr);
    float* dst = isY ? &ldsY[local][0] : &ldsX[local][0];
    float sq = 0.f;
#pragma unroll
    for (int q = 0; q < 8; ++q) {
      float4 v = p[q];
      dst[q * 4 + 0] = v.x; dst[q * 4 + 1] = v.y;
      dst[q * 4 + 2] = v.z; dst[q * 4 + 3] = v.w;
      sq += v.x * v.x + v.y * v.y + v.z * v.z + v.w * v.w;
    }
    if (isY) sqY[local] = sq; else sqX[local] = sq;
  }
  const float c0 = ci[b * 8 + 0], c1 = ci[b * 8 + 1], c2 = ci[b * 8 + 2];
  const float c3 = ci[b * 8 + 3], c4 = ci[b * 8 + 4];
  __syncthreads();

  const int lane  = t & 31;
  const int w     = t >> 5;          // wave 0..3, each does a 16x64 strip
  const int mrow  = lane & 15;       // fragment row (A) / col (B,D)
  const int khalf = (lane >> 4) * 2; // K sub-offset per ISA layout
  const int mBase = w * 16;

  v8f acc0 = {}, acc1 = {}, acc2 = {}, acc3 = {};
#pragma unroll
  for (int kk = 0; kk < KDIM; kk += 4) {
    v2f a  = *(const v2f*)&ldsX[mBase + mrow][kk + khalf];
    v2f b0 = *(const v2f*)&ldsY[ 0 + mrow][kk + khalf];
    v2f b1 = *(const v2f*)&ldsY[16 + mrow][kk + khalf];
    v2f b2 = *(const v2f*)&ldsY[32 + mrow][kk + khalf];
    v2f b3 = *(const v2f*)&ldsY[48 + mrow][kk + khalf];
    acc0 = __builtin_amdgcn_wmma_f32_16x16x4_f32(false, a, false, b0, (short)0, acc0, false, false);
    acc1 = __builtin_amdgcn_wmma_f32_16x16x4_f32(false, a, false, b1, (short)0, acc1, false, false);
    acc2 = __builtin_amdgcn_wmma_f32_16x16x4_f32(false, a, false, b2, (short)0, acc2, false, false);
    acc3 = __builtin_amdgcn_wmma_f32_16x16x4_f32(false, a, false, b3, (short)0, acc3, false, false);
  }

  // Epilogue: L2 = sq_i + sq_j - 2*gram, 5-bandwidth exp sum.
  float sqA[8];
#pragma unroll
  for (int r = 0; r < 8; ++r) sqA[r] = sqX[mBase + (lane >> 4) * 8 + r];

  float ksum = 0.f;
  {
    float sb = sqY[ 0 + mrow];
#pragma unroll
    for (int r = 0; r < 8; ++r) ksum += kern5(sqA[r] + sb - 2.f * acc0[r], c0, c1, c2, c3, c4);
  }
  {
    float sb = sqY[16 + mrow];
#pragma unroll
    for (int r = 0; r < 8; ++r) ksum += kern5(sqA[r] + sb - 2.f * acc1[r], c0, c1, c2, c3, c4);
  }
  {
    float sb = sqY[32 + mrow];
#pragma unroll
    for (int r = 0; r < 8; ++r) ksum += kern5(sqA[r] + sb - 2.f * acc2[r], c0, c1, c2, c3, c4);
  }
  {
    float sb = sqY[48 + mrow];
#pragma unroll
    for (int r = 0; r < 8; ++r) ksum += kern5(sqA[r] + sb - 2.f * acc3[r], c0, c1, c2, c3, c4);
  }

  // wave32 reduction (fixed order -> deterministic)
#pragma unroll
  for (int off = 16; off > 0; off >>= 1) ksum += __shfl_xor(ksum, off, 32);
  if (lane == 0) redBuf[w] = ksum;
  __syncthreads();
  if (t == 0) {
    tileSum[((size_t)b * TGRID + ti) * TGRID + tj] =
        redBuf[0] + redBuf[1] + redBuf[2] + redBuf[3];
  }
}

// ---------------- Pass 3: deterministic quadrant combine ----------------
__global__ void mmd_pass3(const float* __restrict__ tileSum,
                          float* __restrict__ out) {
  const int b = blockIdx.x;
  const int t = threadIdx.x;         // 256
  float acc = 0.f;
  for (int p = t; p < TGRID * TGRID; p += 256) {
    int ti = p >> 6;
    int tj = p & 63;
    int lo = (ti < tj) ? ti : tj;
    int hi = (ti < tj) ? tj : ti;    // mirror lower triangle (symmetric kernel)
    float S = tileSum[((size_t)b * TGRID + lo) * TGRID + hi];
    int qi = (ti >= QTILES), qj = (tj >= QTILES);
    acc += (qi == qj) ? S : -S;      // XX + YY - XY - YX
  }
  __shared__ float red[256];
  red[t] = acc;
  __syncthreads();
  for (int off = 128; off > 0; off >>= 1) {
    if (t < off) red[t] += red[t + off];
    __syncthreads();
  }
  if (t == 0) out[b] = red[0] * (1.0f / ((float)NSIDE * (float)NSIDE));
}

// ---------------- launch ----------------
extern "C" void kernel_launch(void* const* d_in, const int* in_sizes, int n_in,
                              void* d_out, int out_size, void* d_ws, size_t ws_size,
                              hipStream_t stream) {
  const float* src = (const float*)d_in[0];
  const float* tgt = (const float*)d_in[1];
  float* out = (float*)d_out;
  float* ws  = (float*)d_ws;
  float* ci      = ws;        // 32 floats
  float* tileSum = ws + 32;   // 4*64*64 floats (~64KB)

  mmd_pass1<<<NBATCH, 256, 0, stream>>>(src, tgt, ci);
  dim3 g2(TGRID, TGRID, NBATCH);
  mmd_pass2<<<g2, 128, 0, stream>>>(src, tgt, ci, tileSum);
  mmd_pass3<<<NBATCH, 256, 0, stream>>>(tileSum, out);
}